// BanditPrototypeManager_88115549045394
// MI455X (gfx1250) — compile-verified
//
#include <hip/hip_runtime.h>
#include <hip/hip_bf16.h>
#include <math.h>

typedef __attribute__((ext_vector_type(2))) float v2f;
typedef __attribute__((ext_vector_type(8))) float v8f;
typedef unsigned int u32x4 __attribute__((ext_vector_type(4)));
typedef int i32x4 __attribute__((ext_vector_type(4)));
typedef int i32x8 __attribute__((ext_vector_type(8)));

#define B_   4
#define N_   8
#define C_   256
#define K_   16
#define P_   4096          // H*W
#define BN_  32            // B*N
#define CPAD 260           // LDS row pad: stride%64==4 -> conflict-free A reads

#if defined(__has_builtin)
#  if __has_builtin(__builtin_amdgcn_tensor_load_to_lds) && \
      __has_builtin(__builtin_amdgcn_s_wait_tensorcnt)
#    define HAVE_TDM 1
#  endif
#endif

// ---------- helpers ----------
__device__ __forceinline__ float wave_sum(float x) {
#pragma unroll
  for (int o = 16; o > 0; o >>= 1) x += __shfl_xor(x, o, 32);
  return x;
}
__device__ __forceinline__ float wave_max(float x) {
#pragma unroll
  for (int o = 16; o > 0; o >>= 1) x = fmaxf(x, __shfl_xor(x, o, 32));
  return x;
}
__device__ __forceinline__ float gelu_tanh(float x) {
  const float k0 = 0.7978845608028654f; // sqrt(2/pi)
  float t = tanhf(k0 * (x + 0.044715f * x * x * x));
  return 0.5f * x * (1.0f + t);
}

// ---------- kernel 1: masked pooling numerators per (b,n,c) ----------
// b128 streaming loads: at 23.3 TB/s the issue rate of b32 loads limits
// before bandwidth does; float4 keeps the VMEM pipe fed.
__global__ __launch_bounds__(256) void masked_pool_kernel(
    const float* __restrict__ value, const float* __restrict__ mask,
    float* __restrict__ candv, float* __restrict__ candf) {
  __shared__ float red[16];
  int row = blockIdx.x;               // bn*C + c
  int bn  = row >> 8;
  const float4* v4 = (const float4*)(value + (size_t)row * P_);
  const float4* m4 = (const float4*)(mask + (size_t)bn * P_);
  float vm = 0.f, vs = 0.f;
  for (int p = threadIdx.x; p < P_ / 4; p += 256) {
    float4 vv = v4[p];
    float4 mm = m4[p];
    vm += vv.x * mm.x + vv.y * mm.y + vv.z * mm.z + vv.w * mm.w;
    vs += vv.x + vv.y + vv.z + vv.w;
  }
  vm = wave_sum(vm); vs = wave_sum(vs);
  int wave = threadIdx.x >> 5, lane = threadIdx.x & 31;
  if (lane == 0) { red[wave] = vm; red[8 + wave] = vs; }
  __syncthreads();
  if (threadIdx.x == 0) {
    float a = 0.f, b = 0.f;
    for (int i = 0; i < 8; ++i) { a += red[i]; b += red[8 + i]; }
    candv[row] = a; candf[row] = b;
  }
}

// ---------- kernel 2: generic row sum (mask denoms, frame g_t) ----------
__global__ __launch_bounds__(256) void row_sum_kernel(
    const float* __restrict__ in, float* __restrict__ out) {
  __shared__ float red[8];
  const float4* r4 = (const float4*)(in + (size_t)blockIdx.x * P_);
  float s = 0.f;
  for (int p = threadIdx.x; p < P_ / 4; p += 256) {
    float4 vv = r4[p];
    s += vv.x + vv.y + vv.z + vv.w;
  }
  s = wave_sum(s);
  int wave = threadIdx.x >> 5, lane = threadIdx.x & 31;
  if (lane == 0) red[wave] = s;
  __syncthreads();
  if (threadIdx.x == 0) {
    float a = 0.f;
    for (int i = 0; i < 8; ++i) a += red[i];
    out[blockIdx.x] = a;
  }
}

// ---------- kernel 3: policy / bank update / MLP logits ----------
__global__ __launch_bounds__(256) void policy_kernel(
    const float* __restrict__ candv, const float* __restrict__ candf,
    const float* __restrict__ msum,  const float* __restrict__ gtsum,
    const float* __restrict__ proto, const float* __restrict__ age,
    const float* __restrict__ usage, const float* __restrict__ conf,
    const unsigned char* __restrict__ valid,
    const float* __restrict__ W1, const float* __restrict__ b1,
    const float* __restrict__ W2, const float* __restrict__ b2,
    float* __restrict__ pnew, float* __restrict__ vnew,
    float* __restrict__ logits) {
  __shared__ float candL[8][C_];
  __shared__ float hidL[8][C_];
  __shared__ float simL[8][K_];
  __shared__ float red[16];
  __shared__ float agemax_s, usemax_s;
  int tid = threadIdx.x, wave = tid >> 5, lane = tid & 31;

  // global max of age / usage
  float am = 0.f, um = 0.f;
  for (int i = tid; i < BN_ * K_; i += 256) {
    am = fmaxf(am, age[i]); um = fmaxf(um, usage[i]);
  }
  am = wave_max(am); um = wave_max(um);
  if (lane == 0) { red[wave] = am; red[8 + wave] = um; }
  __syncthreads();
  if (tid == 0) {
    float a = 0.f, u = 0.f;
    for (int i = 0; i < 8; ++i) { a = fmaxf(a, red[i]); u = fmaxf(u, red[8 + i]); }
    agemax_s = a; usemax_s = u;
  }
  __syncthreads();
  float agemax = fmaxf(agemax_s, 1.0f), usemax = fmaxf(usemax_s, 1.0f);

  for (int pr = wave; pr < BN_; pr += 8) {   // one (b,n) pair per wave
    int b = pr >> 3;
    float denom = msum[pr];
    bool use_fb = (denom <= 1e-5f);
    float dinv = 1.0f / fmaxf(denom, 1e-6f);

    // candidate prototype (l2 normalized)
    float n2 = 0.f;
#pragma unroll
    for (int i = 0; i < 8; ++i) {
      int c = lane + i * 32;
      float cd = use_fb ? (candf[pr * C_ + c] * (1.0f / P_))
                        : (candv[pr * C_ + c] * dinv);
      candL[wave][c] = cd;
      n2 += cd * cd;
    }
    n2 = wave_sum(n2);
    float cinv = 1.0f / fmaxf(sqrtf(n2), 1e-12f);
#pragma unroll
    for (int i = 0; i < 8; ++i) candL[wave][lane + i * 32] *= cinv;

    // cosine sim vs normalized bank
    for (int k = 0; k < K_; ++k) {
      const float* pk = proto + ((size_t)pr * K_ + k) * C_;
      float d = 0.f, pn = 0.f;
#pragma unroll
      for (int i = 0; i < 8; ++i) {
        int c = lane + i * 32;
        float pv = pk[c];
        d += pv * candL[wave][c];
        pn += pv * pv;
      }
      d = wave_sum(d); pn = wave_sum(pn);
      float s = d / fmaxf(sqrtf(pn), 1e-12f);
      simL[wave][k] = (valid[pr * K_ + k] != 0) ? s : -1.0f;
    }

    // scalar slot/action logic (uniform across lanes)
    bool anyv = false, hasfree = false;
    int tslot = 0, freeslot = -1, worst = 0;
    float msim = -2.0f, wbest = -1e30f;
    for (int k = 0; k < K_; ++k) {
      float s = simL[wave][k];
      if (s > msim) { msim = s; tslot = k; }
      bool vld = valid[pr * K_ + k] != 0;
      anyv |= vld;
      if (!vld && freeslot < 0) freeslot = k;
      hasfree |= !vld;
      float wsc = age[pr * K_ + k] - usage[pr * K_ + k] - conf[pr * K_ + k] +
                  (vld ? 0.f : 1000.f);
      if (wsc > wbest) { wbest = wsc; worst = k; }
    }
    if (!anyv) { tslot = 0; msim = 0.0f; }
    int action = 1;                       // REFINE
    if (msim > 0.85f) action = 0;         // KEEP
    bool low = msim < 0.5f;
    if (low && hasfree)  action = 3;      // SPAWN
    if (low && !hasfree) action = 2;      // REPLACE
    bool refine = (action == 1), replace = (action == 2), spawn = (action == 3);
    bool update = refine || replace || spawn;
    bool rlike  = replace || spawn;
    int chosen = tslot;
    if (spawn)   chosen = hasfree ? freeslot : worst;
    if (replace) chosen = worst;

    // norm of refined blend
    const float* pch = proto + ((size_t)pr * K_ + chosen) * C_;
    float bn2 = 0.f;
#pragma unroll
    for (int i = 0; i < 8; ++i) {
      int c = lane + i * 32;
      float bl = 0.9f * pch[c] + 0.1f * candL[wave][c];
      bn2 += bl * bl;
    }
    bn2 = wave_sum(bn2);
    float binv = 1.0f / fmaxf(sqrtf(bn2), 1e-12f);

    // scatter-updated proto bank + validity (only state needed downstream)
    for (int k = 0; k < K_; ++k) {
      const float* pk = proto + ((size_t)pr * K_ + k) * C_;
      float* po = pnew + ((size_t)pr * K_ + k) * C_;
#pragma unroll
      for (int i = 0; i < 8; ++i) {
        int c = lane + i * 32;
        float op = pk[c], ov = op;
        if (k == chosen) {
          if (refine)     ov = (0.9f * op + 0.1f * candL[wave][c]) * binv;
          else if (rlike) ov = candL[wave][c];
        }
        po[c] = ov;
      }
    }
    if (lane < K_) {
      bool vld = valid[pr * K_ + lane] != 0;
      vnew[pr * K_ + lane] = (vld || (lane == chosen && update)) ? 1.0f : 0.0f;
    }

    // policy MLP: context = [g_t(256) | sim(16) | stats(48)] -> gelu(.)W2+b2
#pragma unroll
    for (int i = 0; i < 8; ++i) {
      int hh = lane + i * 32;
      float acc = b1[hh];
      for (int c = 0; c < C_; ++c)
        acc += (gtsum[b * C_ + c] * (1.0f / P_)) * W1[c * 256 + hh];
      for (int k = 0; k < K_; ++k)
        acc += simL[wave][k] * W1[(C_ + k) * 256 + hh];
      for (int k = 0; k < K_; ++k) {
        int base = (C_ + K_ + 3 * k) * 256 + hh;
        acc += (age[pr * K_ + k] / agemax)   * W1[base];
        acc += (usage[pr * K_ + k] / usemax) * W1[base + 256];
        acc += conf[pr * K_ + k]             * W1[base + 512];
      }
      hidL[wave][hh] = gelu_tanh(acc);
    }
    for (int o = 0; o < 4; ++o) {
      float a = 0.f;
#pragma unroll
      for (int i = 0; i < 8; ++i) {
        int hh = lane + i * 32;
        a += hidL[wave][hh] * W2[hh * 4 + o];
      }
      a = wave_sum(a);
      if (lane == 0) logits[pr * 4 + o] = a + b2[o];
    }
  }
}

// ---------- kernel 4: WMMA readout + fused conditioning ----------
// grid: BN_ * 32 blocks (each block = 128 pixels), 256 threads = 8 waves,
// each wave owns one 16-pixel tile. proto tile staged via the Tensor Data
// Mover (TDM) with hardware row-padding to the CPAD layout.
__global__ __launch_bounds__(256) void readout_kernel(
    const float* __restrict__ value, const float* __restrict__ frame,
    const float* __restrict__ protoN, const float* __restrict__ validN,
    const float* __restrict__ pgate, const float* __restrict__ fgate,
    float* __restrict__ out) {
  __shared__ float pL[K_ * CPAD];   // proto_new   [slot][c], CPAD-padded rows
  __shared__ float nbL[K_ * CPAD];  // normalized  [slot][c]
  __shared__ float wL[8 * K_ * 17]; // softmax wts [wave][slot][pix], padded
  __shared__ float invn[K_];
  __shared__ float vmask[K_];
  __shared__ float anyvS;

  int bn = blockIdx.x >> 5;
  int pblk = blockIdx.x & 31;
  int tid = threadIdx.x;

  const float* pbase = protoN + (size_t)bn * K_ * C_;
#ifdef HAVE_TDM
  // TDM: DMA the 16x256 f32 tile global->LDS; pad 4 DWORDs after every
  // 256 DWORDs so LDS rows land on the CPAD=260 stride.
  if (tid < 32) {
    unsigned long long ga = (unsigned long long)pbase;
    unsigned ldsOff = (unsigned)(unsigned long long)(const void*)pL;
    u32x4 g0 = { 1u,                                    // count=1, user mode
                 ldsOff,                                // lds_addr
                 (unsigned)(ga & 0xffffffffu),          // global_addr[31:0]
                 (unsigned)(((ga >> 32) & 0x1ffffffu)   // global_addr[56:32]
                            | (2u << 30)) };            // type=2 (image)
    i32x8 g1 = { (int)((2u << 16)      // data_size=4B
                     | (1u << 20)      // pad_enable
                     | (7u << 22)      // pad_interval: 256 DWORDs
                     | (3u << 25)),    // pad_amount: 4 DWORDs
                 (int)(0x0100u << 16), // tensor_dim0 lo16 = 256
                 (int)(16u << 16),     // dim0 hi=0 | tensor_dim1 lo16 = 16
                 (int)(256u << 16),    // dim1 hi=0 | tile_dim0 = 256
                 16,                   // tile_dim1 = 16 | tile_dim2 = 0
                 256,                  // tensor_dim0_stride lo32 = 256
                 0, 0 };
    i32x4 gz = { 0, 0, 0, 0 };
#if __clang_major__ >= 23
    i32x8 gz8 = { 0, 0, 0, 0, 0, 0, 0, 0 };
    __builtin_amdgcn_tensor_load_to_lds(g0, g1, gz, gz, gz8, 0);
#else
    __builtin_amdgcn_tensor_load_to_lds(g0, g1, gz, gz, 0);
#endif
    __builtin_amdgcn_s_wait_tensorcnt(0);  // issuing wave waits; barrier
  }                                        // below publishes to all waves
#else
  for (int i = tid; i < K_ * C_; i += 256) {
    int k = i >> 8, c = i & 255;
    pL[k * CPAD + c] = pbase[i];
  }
#endif
  __syncthreads();
  if (tid < K_) {
    float s = 0.f;
    for (int c = 0; c < C_; ++c) { float v = pL[tid * CPAD + c]; s += v * v; }
    invn[tid]  = 1.0f / fmaxf(sqrtf(s), 1e-12f);
    vmask[tid] = validN[bn * K_ + tid];
  }
  __syncthreads();
  for (int i = tid; i < K_ * C_; i += 256) {
    int k = i >> 8, c = i & 255;
    nbL[k * CPAD + c] = pL[k * CPAD + c] * invn[k];
  }
  if (tid == 0) {
    float a = 0.f;
    for (int k = 0; k < K_; ++k) a = fmaxf(a, vmask[k]);
    anyvS = a;
  }
  __syncthreads();

  int wave = tid >> 5, lane = tid & 31;
  int m = lane & 15, half = lane >> 4, koff = half * 2;
  int p0 = pblk * 128 + wave * 16;
  const float* vb = value + (size_t)bn * C_ * P_;
  const float* fb = frame + (size_t)(bn >> 3) * C_ * P_;
  float pg = pgate[0], fg = fgate[0];
  float anyv = anyvS;

  // --- scores[k,p] = nb[k,:]·v[:,p] via WMMA f32 16x16x4 (64 steps over C) ---
  v8f acc = {};
  float s2 = 0.f;
#pragma unroll 4
  for (int c0 = 0; c0 < C_; c0 += 4) {
    v2f a, bfrag;
    a.x = nbL[m * CPAD + c0 + koff];
    a.y = nbL[m * CPAD + c0 + koff + 1];
    float v0 = vb[(size_t)(c0 + koff) * P_ + p0 + m];
    float v1 = vb[(size_t)(c0 + koff + 1) * P_ + p0 + m];
    bfrag.x = v0; bfrag.y = v1;
    s2 += v0 * v0 + v1 * v1;  // per-pixel ||v||^2 (lane-pair split over c)
    acc = __builtin_amdgcn_wmma_f32_16x16x4_f32(false, a, false, bfrag,
                                                (short)0, acc, false, false);
  }
  float s2t = s2 + __shfl_xor(s2, 16, 32);
  float inv_vn = 1.0f / fmaxf(sqrtf(s2t), 1e-12f);

  // --- softmax over K (slots 0-7 / 8-15 live in lane halves) ---
  float sc[8];
  float lmax = -1e30f;
#pragma unroll
  for (int r = 0; r < 8; ++r) {
    int slot = r + half * 8;
    float s = acc[r] * inv_vn;
    s = (vmask[slot] > 0.5f) ? s : -1e9f;
    sc[r] = s;
    lmax = fmaxf(lmax, s);
  }
  float gmax = fmaxf(lmax, __shfl_xor(lmax, 16, 32));
  float lsum = 0.f;
#pragma unroll
  for (int r = 0; r < 8; ++r) { sc[r] = __expf(sc[r] - gmax); lsum += sc[r]; }
  float gsum = lsum + __shfl_xor(lsum, 16, 32);
  float rs = 1.0f / gsum;
  float* wW = wL + wave * K_ * 17;
#pragma unroll
  for (int r = 0; r < 8; ++r) {
    int slot = r + half * 8;
    wW[slot * 17 + m] = sc[r] * rs;
  }

  // --- proto_map[c,p] = sum_k proto[k,c]*w[k,p] + fused conditioning ---
  float gate = pg * anyv;
  for (int ct = 0; ct < 16; ++ct) {
    int cb = ct * 16;
    v8f pm = {};
#pragma unroll
    for (int k0 = 0; k0 < 16; k0 += 4) {
      v2f a, bfrag;
      a.x = pL[(k0 + koff) * CPAD + cb + m];       // A[m=c_row][k=slot]
      a.y = pL[(k0 + koff + 1) * CPAD + cb + m];
      bfrag.x = wW[(k0 + koff) * 17 + m];          // B[k=slot][n=pixel]
      bfrag.y = wW[(k0 + koff + 1) * 17 + m];
      pm = __builtin_amdgcn_wmma_f32_16x16x4_f32(false, a, false, bfrag,
                                                 (short)0, pm, false, false);
    }
#pragma unroll
    for (int r = 0; r < 8; ++r) {
      int c = cb + r + half * 8;
      size_t gi = (size_t)c * P_ + p0 + m;
      float vv = vb[gi];   // re-touches score-pass working set: WGP$/L2 hit
      float ff = fb[gi];
      out[(size_t)bn * C_ * P_ + gi] = vv + gate * pm[r] + fg * ff;
    }
  }
}

// ---------- host launcher ----------
extern "C" void kernel_launch(void* const* d_in, const int* in_sizes, int n_in,
                              void* d_out, int out_size, void* d_ws, size_t ws_size,
                              hipStream_t stream) {
  (void)in_sizes; (void)n_in; (void)out_size; (void)ws_size;
  const float* value = (const float*)d_in[0];
  const float* frame = (const float*)d_in[1];
  const float* mask  = (const float*)d_in[2];
  const float* proto = (const float*)d_in[3];
  const float* age   = (const float*)d_in[4];
  const float* usage = (const float*)d_in[5];
  const float* conf  = (const float*)d_in[6];
  const unsigned char* valid = (const unsigned char*)d_in[7];
  const float* W1 = (const float*)d_in[8];
  const float* b1 = (const float*)d_in[9];
  const float* W2 = (const float*)d_in[10];
  const float* b2 = (const float*)d_in[11];
  const float* pgate = (const float*)d_in[12];
  const float* fgate = (const float*)d_in[13];

  float* ws    = (float*)d_ws;
  float* candv = ws;                 // [BN,C]      8192
  float* candf = ws + 8192;          // [BN,C]      8192
  float* msum  = ws + 16384;         // [BN]        32
  float* gtsum = ws + 16416;         // [B,C]       1024
  float* pnew  = ws + 17440;         // [BN,K,C]    131072
  float* vnew  = ws + 50208;         // [BN,K]      512

  float* cond   = (float*)d_out;
  float* logits = cond + (size_t)BN_ * C_ * P_;

  masked_pool_kernel<<<BN_ * C_, 256, 0, stream>>>(value, mask, candv, candf);
  row_sum_kernel<<<B_ * C_, 256, 0, stream>>>(frame, gtsum);
  row_sum_kernel<<<BN_, 256, 0, stream>>>(mask, msum);
  policy_kernel<<<1, 256, 0, stream>>>(candv, candf, msum, gtsum, proto, age,
                                       usage, conf, valid, W1, b1, W2, b2,
                                       pnew, vnew, logits);
  readout_kernel<<<BN_ * 32, 256, 0, stream>>>(value, frame, pnew, vnew,
                                               pgate, fgate, cond);
}